// BiLSTMClassifier_64364379898007
// MI455X (gfx1250) — compile-verified
//
#include <hip/hip_runtime.h>
#include <hip/hip_bf16.h>
#include <math.h>

typedef float v2f __attribute__((ext_vector_type(2)));
typedef float v8f __attribute__((ext_vector_type(8)));

#define T_LEN 512
#define E_DIM 64
#define H_DIM 64
#define G_DIM 256   // 4*H
#define C_DIM 8
#define B_SZ  512
#define ROWS  16    // batch rows per workgroup (= WMMA M)
#define XS    66    // padded LDS stride for 16x64 matrices (bank-conflict-free, 8B aligned)
#define ZS    260   // padded LDS stride for z (16x256)

__device__ __forceinline__ float sigf(float x) { return 1.0f / (1.0f + __expf(-x)); }

__launch_bounds__(512, 1)
__global__ void bilstm_fused_kernel(
    const int* __restrict__ ids,
    const int* __restrict__ positions,
    const unsigned char* __restrict__ amask,
    const float* __restrict__ word_table,
    const float* __restrict__ pos_table,
    const float* __restrict__ Wx,
    const float* __restrict__ Wh,
    const float* __restrict__ bvec,
    const float* __restrict__ Wd,
    const float* __restrict__ bd,
    float* __restrict__ out)
{
    __shared__ float s_x [ROWS*XS];
    __shared__ float s_h1[ROWS*XS];
    __shared__ float s_c1[ROWS*XS];
    __shared__ float s_h2[ROWS*XS];
    __shared__ float s_c2[ROWS*XS];
    __shared__ float s_z [ROWS*ZS];
    __shared__ float s_wd[H_DIM*C_DIM];
    __shared__ float s_bd[C_DIM];
    __shared__ float s_logit[ROWS*C_DIM];
    __shared__ unsigned char s_m[ROWS];

    const int tid  = threadIdx.x;
    const int w    = tid >> 5;        // wave id == N-tile (0..15)
    const int lane = tid & 31;
    const int half = lane >> 4;
    const int m    = lane & 15;
    const int col  = w * 16 + m;      // z / weight column owned by this lane
    const int b0   = blockIdx.x * ROWS;

    // ---------------- one-time init ----------------
    for (int i = tid; i < ROWS*XS; i += 512) {
        s_h1[i] = 0.f; s_c1[i] = 0.f; s_h2[i] = 0.f; s_c2[i] = 0.f;
    }
    s_wd[tid] = Wd[tid];              // H*C == 512 == blockDim
    if (tid < C_DIM) s_bd[tid] = bd[tid];

    // Preload B-operand fragments of Wx / Wh into registers.
    // B 4x16 fragment: lanes 0-15 hold K=(4kk+0,4kk+1), lanes 16-31 K=(4kk+2,4kk+3), N = lane%16.
    v2f bwx[16], bwh[16];
    #pragma unroll
    for (int kk = 0; kk < 16; ++kk) {
        const int kr = 4*kk + 2*half;
        bwx[kk].x = Wx[(kr    )*G_DIM + col];
        bwx[kk].y = Wx[(kr + 1)*G_DIM + col];
        bwh[kk].x = Wh[(kr    )*G_DIM + col];
        bwh[kk].y = Wh[(kr + 1)*G_DIM + col];
    }
    const float bias = bvec[col];

    // Embedding prefetch for t = 0 (each thread covers 2 of 16x64 elements).
    const int r0 = tid >> 6, e0 = tid & 63;
    const int r1 = (tid + 512) >> 6;
    float xr0, xr1; unsigned char mr = 1;
    {
        const int ia = ids[(b0 + r0)*T_LEN];
        const int ib = ids[(b0 + r1)*T_LEN];
        const int pa = positions[(b0 + r0)*T_LEN];
        const int pb = positions[(b0 + r1)*T_LEN];
        xr0 = word_table[ia*E_DIM + e0] + pos_table[pa*E_DIM + e0];
        xr1 = word_table[ib*E_DIM + e0] + pos_table[pb*E_DIM + e0];
        if (tid < ROWS) mr = amask[(b0 + tid)*T_LEN];
    }

    for (int t = 0; t < T_LEN; ++t) {
        // stage prefetched embedding into LDS
        s_x[r0*XS + e0] = xr0;
        s_x[r1*XS + e0] = xr1;
        if (tid < ROWS) s_m[tid] = mr;
        __syncthreads();

        // issue next-step gather early; latency overlaps the WMMA work below
        if (t + 1 < T_LEN) {
            const int ia = ids[(b0 + r0)*T_LEN + t + 1];
            const int ib = ids[(b0 + r1)*T_LEN + t + 1];
            const int pa = positions[(b0 + r0)*T_LEN + t + 1];
            const int pb = positions[(b0 + r1)*T_LEN + t + 1];
            xr0 = word_table[ia*E_DIM + e0] + pos_table[pa*E_DIM + e0];
            xr1 = word_table[ib*E_DIM + e0] + pos_table[pb*E_DIM + e0];
            if (tid < ROWS) mr = amask[(b0 + tid)*T_LEN + t + 1];
        }

        // ---------------- layer 1: z = b + x@Wx + h1@Wh ----------------
        {
            v8f acc0, acc1;
            #pragma unroll
            for (int j = 0; j < 8; ++j) { acc0[j] = bias; acc1[j] = 0.f; }
            #pragma unroll
            for (int kk = 0; kk < 16; kk += 2) {
                const int kra = 4*kk + 2*half, krb = 4*(kk+1) + 2*half;
                v2f a0 = *(const v2f*)&s_x[m*XS + kra];
                v2f a1 = *(const v2f*)&s_x[m*XS + krb];
                acc0 = __builtin_amdgcn_wmma_f32_16x16x4_f32(false, a0, false, bwx[kk  ], (short)0, acc0, false, false);
                acc1 = __builtin_amdgcn_wmma_f32_16x16x4_f32(false, a1, false, bwx[kk+1], (short)0, acc1, false, false);
            }
            #pragma unroll
            for (int kk = 0; kk < 16; kk += 2) {
                const int kra = 4*kk + 2*half, krb = 4*(kk+1) + 2*half;
                v2f a0 = *(const v2f*)&s_h1[m*XS + kra];
                v2f a1 = *(const v2f*)&s_h1[m*XS + krb];
                acc0 = __builtin_amdgcn_wmma_f32_16x16x4_f32(false, a0, false, bwh[kk  ], (short)0, acc0, false, false);
                acc1 = __builtin_amdgcn_wmma_f32_16x16x4_f32(false, a1, false, bwh[kk+1], (short)0, acc1, false, false);
            }
            #pragma unroll
            for (int j = 0; j < 8; ++j)
                s_z[(j + 8*half)*ZS + col] = acc0[j] + acc1[j];
        }
        __syncthreads();

        // ---------------- gates layer 1 ----------------
        #pragma unroll
        for (int q = 0; q < 2; ++q) {
            const int idx = tid + q*512;
            const int r = idx >> 6, jc = idx & 63;
            const float zi = s_z[r*ZS + jc      ];
            const float zf = s_z[r*ZS + 64 + jc ];
            const float zg = s_z[r*ZS + 128 + jc];
            const float zo = s_z[r*ZS + 192 + jc];
            const float cold = s_c1[r*XS + jc], hold = s_h1[r*XS + jc];
            const float cn = sigf(zf)*cold + sigf(zi)*tanhf(zg);
            const float hn = sigf(zo)*tanhf(cn);
            const bool keep = (s_m[r] != 0);
            s_c1[r*XS + jc] = keep ? cn : cold;
            s_h1[r*XS + jc] = keep ? hn : hold;
        }
        __syncthreads();

        // ---------------- layer 2: z = b + h1@Wx + h2@Wh (weights shared) ----------------
        {
            v8f acc0, acc1;
            #pragma unroll
            for (int j = 0; j < 8; ++j) { acc0[j] = bias; acc1[j] = 0.f; }
            #pragma unroll
            for (int kk = 0; kk < 16; kk += 2) {
                const int kra = 4*kk + 2*half, krb = 4*(kk+1) + 2*half;
                v2f a0 = *(const v2f*)&s_h1[m*XS + kra];
                v2f a1 = *(const v2f*)&s_h1[m*XS + krb];
                acc0 = __builtin_amdgcn_wmma_f32_16x16x4_f32(false, a0, false, bwx[kk  ], (short)0, acc0, false, false);
                acc1 = __builtin_amdgcn_wmma_f32_16x16x4_f32(false, a1, false, bwx[kk+1], (short)0, acc1, false, false);
            }
            #pragma unroll
            for (int kk = 0; kk < 16; kk += 2) {
                const int kra = 4*kk + 2*half, krb = 4*(kk+1) + 2*half;
                v2f a0 = *(const v2f*)&s_h2[m*XS + kra];
                v2f a1 = *(const v2f*)&s_h2[m*XS + krb];
                acc0 = __builtin_amdgcn_wmma_f32_16x16x4_f32(false, a0, false, bwh[kk  ], (short)0, acc0, false, false);
                acc1 = __builtin_amdgcn_wmma_f32_16x16x4_f32(false, a1, false, bwh[kk+1], (short)0, acc1, false, false);
            }
            #pragma unroll
            for (int j = 0; j < 8; ++j)
                s_z[(j + 8*half)*ZS + col] = acc0[j] + acc1[j];
        }
        __syncthreads();

        // ---------------- gates layer 2 ----------------
        #pragma unroll
        for (int q = 0; q < 2; ++q) {
            const int idx = tid + q*512;
            const int r = idx >> 6, jc = idx & 63;
            const float zi = s_z[r*ZS + jc      ];
            const float zf = s_z[r*ZS + 64 + jc ];
            const float zg = s_z[r*ZS + 128 + jc];
            const float zo = s_z[r*ZS + 192 + jc];
            const float cold = s_c2[r*XS + jc], hold = s_h2[r*XS + jc];
            const float cn = sigf(zf)*cold + sigf(zi)*tanhf(zg);
            const float hn = sigf(zo)*tanhf(cn);
            const bool keep = (s_m[r] != 0);
            s_c2[r*XS + jc] = keep ? cn : cold;
            s_h2[r*XS + jc] = keep ? hn : hold;
        }
        __syncthreads();

        // ---------------- classifier: logits = h2 @ Wd + bd ----------------
        if (tid < ROWS*C_DIM) {
            const int r = tid >> 3, cc = tid & 7;
            float acc = s_bd[cc];
            #pragma unroll
            for (int k = 0; k < H_DIM; ++k)
                acc += s_h2[r*XS + k] * s_wd[k*C_DIM + cc];
            s_logit[tid] = acc;
        }
        __syncthreads();

        // ---------------- softmax + store (fwd and duplicated bwd half) ----------------
        if (tid < ROWS) {
            const int r = tid;
            float l[C_DIM];
            float mx = -INFINITY;
            #pragma unroll
            for (int c = 0; c < C_DIM; ++c) { l[c] = s_logit[r*C_DIM + c]; mx = fmaxf(mx, l[c]); }
            float s = 0.f;
            #pragma unroll
            for (int c = 0; c < C_DIM; ++c) { l[c] = __expf(l[c] - mx); s += l[c]; }
            const float inv = 1.0f / s;
            const size_t base = ((size_t)(b0 + r) * (2*T_LEN) + t) * C_DIM;
            #pragma unroll
            for (int c = 0; c < C_DIM; ++c) {
                out[base + c] = l[c] * inv;
                out[base + (size_t)T_LEN * C_DIM + c] = l[c] * inv;
            }
        }
        __syncthreads();
    }
}

extern "C" void kernel_launch(void* const* d_in, const int* in_sizes, int n_in,
                              void* d_out, int out_size, void* d_ws, size_t ws_size,
                              hipStream_t stream) {
    const int*           ids        = (const int*)d_in[0];
    const int*           positions  = (const int*)d_in[1];
    const unsigned char* amask      = (const unsigned char*)d_in[2];
    const float*         word_table = (const float*)d_in[3];
    const float*         pos_table  = (const float*)d_in[4];
    const float*         Wx         = (const float*)d_in[5];
    const float*         Wh         = (const float*)d_in[6];
    const float*         bvec       = (const float*)d_in[7];
    const float*         Wd         = (const float*)d_in[8];
    const float*         bd         = (const float*)d_in[9];
    float*               out        = (float*)d_out;

    dim3 grid(B_SZ / ROWS);   // 32 workgroups, each owns 16 batch rows for the whole scan
    dim3 block(512);          // 16 wave32 = 16 N-tiles of the 256-wide gate matrix
    bilstm_fused_kernel<<<grid, block, 0, stream>>>(
        ids, positions, amask, word_table, pos_table, Wx, Wh, bvec, Wd, bd, out);
}